// WDMMux_4200478015870
// MI455X (gfx1250) — compile-verified
//
#include <hip/hip_runtime.h>

typedef __attribute__((ext_vector_type(2))) float v2f;
typedef __attribute__((ext_vector_type(8))) float v8f;

#define L_IN      524288
#define NUP       2097152
#define NCH       5

/* LDS layout (in floats) — partials alias the filter region after a barrier */
#define HP_STRIDE   2544      /* per-phase padded polyphase filter row (need 2535) */
#define HP_BASE     0         /* 4 * 2544 = 10176 floats                           */
#define SIG_STRIDE  2768      /* per-sequence swizzled signal window (need 2761)   */
#define SIG_BASE    10176     /* 10 * 2768 = 27680 -> end 37856                    */
#define PART_BASE   0         /* 2048 float2 = 4096 floats, aliases hp region      */
#define LDS_FLOATS  37856
#define LDS_BYTES   (LDS_FLOATS * 4)   /* 151424 B -> 2 blocks / WGP */

#define WIN_LO      2504      /* window covers s[jb-2504 .. jb+256] */
#define WIN_N       2761
#define NT0         630       /* t0 = -16 .. 2500 step 4            */

__device__ __forceinline__ int swz(int i) { return i ^ ((i >> 4) & 15); }

/* cos(pi/4 * k) for k in [0,8) — branch-free, matches fp32 reference to <5e-8 */
__device__ __forceinline__ float cos_pi4(int k) {
    const float S = 0.70710678118654752f;
    float mag = (k & 1) ? S : ((k & 2) ? 0.0f : 1.0f);
    return (k >= 3 && k <= 5) ? -mag : mag;
}
__device__ __forceinline__ float sin_pi4(int k) { return cos_pi4((k + 6) & 7); }

__global__ __launch_bounds__(256) void wdm_mux_kernel(
    const float* __restrict__ sre, const float* __restrict__ sim,
    const float* __restrict__ filt, float2* __restrict__ out)
{
    extern __shared__ float lds[];
    const int tid = threadIdx.x;
    const int bi  = blockIdx.x;
    const int jb  = 256 * bi + 1250;

    /* ---- stage polyphase filter into LDS: hp[r][u] = h[r + 4*(u-16)] ---- */
    for (int idx = tid; idx < 4 * HP_STRIDE; idx += 256) {
        int r = idx / HP_STRIDE;
        int u = idx - r * HP_STRIDE;
        int t = u - 16;
        int k = r + 4 * t;
        float v = 0.0f;
        if (t >= 0 && k <= 10000) v = filt[k];
        lds[HP_BASE + idx] = v;
    }

    /* ---- stage 10 signal windows (XOR-swizzled, zero-padded at edges) --- */
    const int j0 = jb - WIN_LO;
    for (int idx = tid; idx < 10 * WIN_N; idx += 256) {
        int seq  = idx / WIN_N;
        int i    = idx - seq * WIN_N;
        int c    = seq >> 1;
        int comp = seq & 1;
        int j    = j0 + i;
        float v  = 0.0f;
        if (j >= 0 && j < L_IN) {
            const float* s = comp ? sim : sre;
            v = s[c * L_IN + j];
        }
        lds[SIG_BASE + seq * SIG_STRIDE + swz(i)] = v;
    }
    __syncthreads();

    /* ---- polyphase FIR as 16x16x4 f32 WMMA GEMM ------------------------- */
    const int lane = tid & 31;
    const int wave = tid >> 5;
    const int r    = wave & 3;   /* phase     */
    const int comp = wave >> 2;  /* 0=re 1=im */
    const int mp   = lane & 15;  /* m for A, p for B */
    const int half = lane >> 4;  /* K pair select    */

    v8f acc[NCH];
    #pragma unroll
    for (int c = 0; c < NCH; ++c) {
        #pragma unroll
        for (int v = 0; v < 8; ++v) acc[c][v] = 0.0f;
    }

    const float* hp = &lds[HP_BASE + r * HP_STRIDE];
    int sbase[NCH];
    #pragma unroll
    for (int c = 0; c < NCH; ++c) sbase[c] = SIG_BASE + (2 * c + comp) * SIG_STRIDE;

    int aidx  = mp + 2 * half;                    /* = 16 + m + 2h + t0, t0=-16 */
    int bidx0 = WIN_LO + 16 * mp + 16 - 2 * half; /* = 2504 + 16p - t0 - 2h     */

    #pragma unroll 2
    for (int it = 0; it < NT0; ++it) {
        v2f a;
        a.x = hp[aidx];
        a.y = hp[aidx + 1];
        #pragma unroll
        for (int c = 0; c < NCH; ++c) {
            v2f b;
            b.x = lds[sbase[c] + swz(bidx0)];
            b.y = lds[sbase[c] + swz(bidx0 - 1)];
            acc[c] = __builtin_amdgcn_wmma_f32_16x16x4_f32(
                false, a, false, b, (short)0, acc[c], false, false);
        }
        aidx  += 4;
        bidx0 -= 4;
    }

    /* ---- modulation weights: wave-uniform, branch-free trig ------------- */
    float w1t[2][NCH], w2t[2][NCH];
    #pragma unroll
    for (int par = 0; par < 2; ++par) {
        #pragma unroll
        for (int c = 0; c < NCH; ++c) {
            int kk   = ((c - 2) * (r + 4 * par)) & 7;   /* (ch*n) mod 8 */
            float cs = cos_pi4(kk), sn = sin_pi4(kk);
            w1t[par][c] = comp ? -sn : cs;   /* contribution to Re(out) */
            w2t[par][c] = comp ?  cs : sn;   /* contribution to Im(out) */
        }
    }

    /* all LDS reads of hp/sig are done -> partials may alias hp region */
    __syncthreads();

    float2* part = (float2*)&lds[PART_BASE];
    #pragma unroll
    for (int v = 0; v < 8; ++v) {
        float re = 0.0f, im = 0.0f;
        #pragma unroll
        for (int c = 0; c < NCH; ++c) {
            float y = acc[c][v];
            re += y * w1t[v & 1][c];
            im += y * w2t[v & 1][c];
        }
        part[(wave << 8) + (v << 5) + lane] = make_float2(re, im);
    }
    __syncthreads();

    /* ---- combine re/im component waves, coalesced float2 writeout ------- */
    const int nbase = bi << 10;
    #pragma unroll
    for (int i = 0; i < 4; ++i) {
        int q  = tid + (i << 8);
        int rr = q & 3;
        int qp = q >> 2;          /* = m + 16*p */
        int m  = qp & 15;
        int p  = qp >> 4;
        int v  = m & 7;
        int l  = p + ((m & 8) << 1);
        float2 A = part[(rr << 8) + (v << 5) + l];
        float2 B = part[((rr + 4) << 8) + (v << 5) + l];
        out[nbase + q] = make_float2(A.x + B.x, A.y + B.y);
    }
}

extern "C" void kernel_launch(void* const* d_in, const int* in_sizes, int n_in,
                              void* d_out, int out_size, void* d_ws, size_t ws_size,
                              hipStream_t stream) {
    (void)in_sizes; (void)n_in; (void)out_size; (void)d_ws; (void)ws_size;
    const float* sre  = (const float*)d_in[0];
    const float* sim  = (const float*)d_in[1];
    const float* filt = (const float*)d_in[2];
    float2* out = (float2*)d_out;

    hipFuncSetAttribute((const void*)wdm_mux_kernel,
                        hipFuncAttributeMaxDynamicSharedMemorySize, LDS_BYTES);
    wdm_mux_kernel<<<dim3(NUP / 1024), dim3(256), LDS_BYTES, stream>>>(sre, sim, filt, out);
}